// LanguageEmbeddingSpaceStructureLoss_56324201119795
// MI455X (gfx1250) — compile-verified
//
#include <hip/hip_runtime.h>

typedef __attribute__((ext_vector_type(2))) float v2f;
typedef __attribute__((ext_vector_type(8))) float v8f;

#define NLANG 2048
#define DIM   64
#define VOCAB 2048

__global__ void k_init(float* ws) {
    if (threadIdx.x == 0) {
        ((unsigned*)ws)[0] = 0u;  // max(map_dist) as uint bits (all values >= 0)
        ws[1] = 0.0f;             // masked |emb_dist - metric| sum
        ws[2] = 0.0f;             // mask count
    }
}

__global__ void k_max(const float* __restrict__ m, float* ws, int n) {
    unsigned* wsu = (unsigned*)ws;
    float v = 0.0f;
    for (int i = blockIdx.x * blockDim.x + threadIdx.x; i < n;
         i += gridDim.x * blockDim.x)
        v = fmaxf(v, m[i]);
    #pragma unroll
    for (int off = 16; off > 0; off >>= 1)
        v = fmaxf(v, __shfl_xor(v, off, 32));
    if ((threadIdx.x & 31) == 0)
        atomicMax(&wsu[0], __float_as_uint(v));  // order-preserving for >=0 floats
}

__global__ void __launch_bounds__(256)
k_pairs(const int* __restrict__ ids, const float* __restrict__ E,
        const float* __restrict__ tree, const float* __restrict__ mapd,
        float* ws)
{
    __shared__ float Ei[64][68];   // +4 pad: row stride 272B = 17 x 16B
    __shared__ float Ej[64][68];
    __shared__ int   idI[64], idJ[64];

    const int t  = threadIdx.x;
    const int i0 = blockIdx.y * 64;
    const int j0 = blockIdx.x * 64;

    // Stage 64x64 f32 tiles of embeddings into LDS (float4 = b128 path)
    #pragma unroll
    for (int s = 0; s < 4; ++s) {
        int flat = t + 256 * s;          // float4 index, 1024 per tile
        int row  = flat >> 4;
        int c4   = (flat & 15) << 2;
        float4 vi = *(const float4*)&E[(size_t)(i0 + row) * DIM + c4];
        float4 vj = *(const float4*)&E[(size_t)(j0 + row) * DIM + c4];
        *(float4*)&Ei[row][c4] = vi;
        *(float4*)&Ej[row][c4] = vj;
    }
    if (t < 64)        idI[t]      = ids[i0 + t];
    else if (t < 128)  idJ[t - 64] = ids[j0 + (t - 64)];
    __syncthreads();

    const float inv_largest = 1.0f / __uint_as_float(((const unsigned*)ws)[0]);

    const int wave = t >> 5;
    const int lane = t & 31;
    const int M    = lane & 15;   // WMMA A-operand: M = lane%16
    const int hi   = lane >> 4;   // K = 2*hi + vgpr_index

    v8f c = {0.f, 0.f, 0.f, 0.f, 0.f, 0.f, 0.f, 0.f};
    v2f bones = {1.0f, 1.0f};     // B = 4x16 all-ones -> D[m,n] = sum_k A[m,k]
    float cnt = 0.0f;

    #pragma unroll 1              // keep VGPR pressure under 256: no cross-subtile unroll
    for (int ssub = 0; ssub < 2; ++ssub) {
        int sub = wave + 8 * ssub;           // 16 subtiles of 16x16, 8 waves x 2
        int si = sub >> 2, sj = sub & 3;
        // lane-private rows: pair p = q*64 + M*4 + K ; di = p%16 ; dj = p/16
        int rowI  = si * 16 + ((M & 3) << 2) + (hi << 1);  // r=0 (r=1 -> +1)
        int baseJ = sj * 16 + (M >> 2);                    // dj = 4*q + M/4

        // Issue all 16 scattered L2 gathers up front; consume after the k-loop
        // so the loads' latency hides under ~1K VALU ops of distance work.
        float tt[4][2], mm[4][2], msk[4][2];
        #pragma unroll
        for (int q = 0; q < 4; ++q) {
            #pragma unroll
            for (int r = 0; r < 2; ++r) {
                int di = ((M & 3) << 2) + (hi << 1) + r;
                int dj = (q << 2) + (M >> 2);
                int ii = idI[si * 16 + di];
                int jj = idJ[sj * 16 + dj];
                size_t off = (size_t)ii * VOCAB + jj;
                tt[q][r]  = tree[off];
                mm[q][r]  = mapd[off];
                msk[q][r] = (ii != jj) ? 1.0f : 0.0f;
            }
        }

        float acc[4][2] = {};
        #pragma unroll 4          // bounded unroll: ILP for dual-issue w/o VGPR blowup
        for (int k = 0; k < DIM; k += 4) {
            float4 a0 = *(const float4*)&Ei[rowI][k];
            float4 a1 = *(const float4*)&Ei[rowI + 1][k];
            #pragma unroll
            for (int q = 0; q < 4; ++q) {
                float4 b = *(const float4*)&Ej[baseJ + 4 * q][k];
                acc[q][0] += fabsf(a0.x - b.x) + fabsf(a0.y - b.y)
                           + fabsf(a0.z - b.z) + fabsf(a0.w - b.w);
                acc[q][1] += fabsf(a1.x - b.x) + fabsf(a1.y - b.y)
                           + fabsf(a1.z - b.z) + fabsf(a1.w - b.w);
            }
        }

        #pragma unroll
        for (int q = 0; q < 4; ++q) {
            v2f a;
            #pragma unroll
            for (int r = 0; r < 2; ++r) {
                float metric = 0.5f * (tt[q][r] + mm[q][r] * inv_largest);
                float v = msk[q][r] * fabsf(acc[q][r] * (1.0f / DIM) - metric);
                cnt += msk[q][r];
                a[r] = v;
            }
            // matrix-unit row-sum accumulate: C += A x ones
            c = __builtin_amdgcn_wmma_f32_16x16x4_f32(
                    false, a, false, bones, (short)0, c, false, false);
        }
    }

    float lane_sum = 0.0f;
    #pragma unroll
    for (int r = 0; r < 8; ++r) lane_sum += c[r];

    #pragma unroll
    for (int off = 16; off > 0; off >>= 1) {
        lane_sum += __shfl_xor(lane_sum, off, 32);
        cnt      += __shfl_xor(cnt, off, 32);
    }
    if (lane == 0) {
        // D columns are 16 identical copies of the row sums -> divide by 16
        atomicAdd(&ws[1], lane_sum * (1.0f / 16.0f));
        atomicAdd(&ws[2], cnt);
    }
}

__global__ void k_final(const float* ws, float* out) {
    if (threadIdx.x == 0) out[0] = ws[1] / ws[2];
}

extern "C" void kernel_launch(void* const* d_in, const int* in_sizes, int n_in,
                              void* d_out, int out_size, void* d_ws, size_t ws_size,
                              hipStream_t stream) {
    const int*   ids  = (const int*)  d_in[0];
    const float* E    = (const float*)d_in[1];
    const float* tree = (const float*)d_in[2];
    const float* mapd = (const float*)d_in[3];
    float* ws  = (float*)d_ws;
    float* out = (float*)d_out;

    k_init<<<1, 32, 0, stream>>>(ws);
    k_max<<<512, 256, 0, stream>>>(mapd, ws, VOCAB * VOCAB);
    dim3 grid(NLANG / 64, NLANG / 64);
    k_pairs<<<grid, 256, 0, stream>>>(ids, E, tree, mapd, ws);
    k_final<<<1, 32, 0, stream>>>(ws, out);
}